// GCN_2353642078895
// MI455X (gfx1250) — compile-verified
//
#include <hip/hip_runtime.h>
#include <hip/hip_bf16.h>

#define NN   50000
#define DIM  128
#define ROW_TILES (NN / 16)   // 3125, exact

typedef __attribute__((ext_vector_type(2))) float v2f;
typedef __attribute__((ext_vector_type(8))) float v8f;

// ---------------------------------------------------------------------------
// Degree / normalization
// ---------------------------------------------------------------------------
__global__ void gcn_init_deg(float* __restrict__ deg) {
    int i = blockIdx.x * blockDim.x + threadIdx.x;
    if (i < NN) deg[i] = 1.0f;                       // self-loop contribution
}

__global__ void gcn_accum_deg(const int* __restrict__ dst, float* __restrict__ deg, int E) {
    int e = blockIdx.x * blockDim.x + threadIdx.x;
    if (e < E) atomicAdd(&deg[dst[e]], 1.0f);
}

__global__ void gcn_make_dinv(const float* __restrict__ deg, float* __restrict__ dinv) {
    int i = blockIdx.x * blockDim.x + threadIdx.x;
    if (i < NN) dinv[i] = rsqrtf(fmaxf(deg[i], 1.0f));
}

// ---------------------------------------------------------------------------
// H = X @ W  via V_WMMA_F32_16X16X4_F32  (X: NNxDIM row-major, W: DIMxDIM)
// One wave computes a 16-row strip (8 output tiles of 16x16).
// W is staged through LDS in two 64-row chunks, padded stride 132 floats.
// ---------------------------------------------------------------------------
__global__ __launch_bounds__(256)
void gcn_gemm_wmma(const float* __restrict__ X, const float* __restrict__ W,
                   float* __restrict__ H) {
    constexpr int LDW = 132;                    // padded LDS row stride (floats)
    __shared__ float sW[64 * LDW];              // 33.8 KB, well under WGP LDS

    const int lane = threadIdx.x & 31;
    const int wave = threadIdx.x >> 5;          // 0..7
    const int rowTile = blockIdx.x * 8 + wave;  // 16-row tile index
    const bool active = rowTile < ROW_TILES;    // wave-uniform -> EXEC all-ones
    const int m0 = rowTile * 16;

    // A frag (16x4 f32): lanes 0-15 hold K=k0+{0,1}, lanes 16-31 K=k0+{2,3}
    const int aRow  = m0 + (lane & 15);
    const int aKoff = (lane >> 4) * 2;
    // B frag (4x16): VGPR v holds K = k0+v+2*(lane>=16), N = n0 + (lane&15)
    const int bKoff = (lane >> 4) * 2;
    const int bN    = lane & 15;

    v8f zero = {0.f, 0.f, 0.f, 0.f, 0.f, 0.f, 0.f, 0.f};
    v8f acc[8];
#pragma unroll
    for (int n = 0; n < 8; ++n) acc[n] = zero;

    for (int kh = 0; kh < 2; ++kh) {
        // Cooperative load of 64 rows of W (8192 floats) as float4s.
#pragma unroll
        for (int i = 0; i < 8; ++i) {
            int t = threadIdx.x + i * 256;      // 0..2047 float4 slots
            int r = t >> 5;                     // 32 float4 per 128-float row
            int c = (t & 31) * 4;
            const float4 w4 = *(const float4*)&W[(kh * 64 + r) * DIM + c];
            *(float4*)&sW[r * LDW + c] = w4;
        }
        __syncthreads();

        if (active) {
            for (int kk = 0; kk < 64; kk += 4) {
                // 8-byte contiguous A load (aKoff is even, kk % 4 == 0)
                v2f a = *(const v2f*)&X[aRow * DIM + kh * 64 + kk + aKoff];
                const int bK = kk + bKoff;
#pragma unroll
                for (int n = 0; n < 8; ++n) {
                    v2f b;
                    b.x = sW[(bK + 0) * LDW + n * 16 + bN];
                    b.y = sW[(bK + 1) * LDW + n * 16 + bN];
                    acc[n] = __builtin_amdgcn_wmma_f32_16x16x4_f32(
                        /*neg_a=*/false, a, /*neg_b=*/false, b,
                        /*c_mod=*/(short)0, acc[n],
                        /*reuse_a=*/false, /*reuse_b=*/false);
                }
            }
        }
        __syncthreads();
    }

    if (active) {
        // C/D layout: VGPR v -> row m0 + v + 8*(lane>=16), col n*16 + (lane&15)
        const int cRow = m0 + ((lane >> 4) * 8);
        const int cCol = lane & 15;
#pragma unroll
        for (int n = 0; n < 8; ++n) {
#pragma unroll
            for (int v = 0; v < 8; ++v) {
                H[(cRow + v) * DIM + n * 16 + cCol] = acc[n][v];
            }
        }
    }
}

// ---------------------------------------------------------------------------
// Aggregation
// ---------------------------------------------------------------------------
__global__ void gcn_zero_agg(float* __restrict__ agg) {
    int i = blockIdx.x * blockDim.x + threadIdx.x;
    if (i < NN * DIM) agg[i] = 0.0f;
}

// One thread handles one (edge, 4-feature) chunk: float4 gather of H[src],
// scale by dinv[src]*dinv[dst], 4 global f32 atomics into agg[dst].
__global__ void gcn_scatter(const int* __restrict__ src, const int* __restrict__ dst,
                            const float* __restrict__ dinv, const float* __restrict__ H,
                            float* __restrict__ agg, int E) {
    int idx = blockIdx.x * blockDim.x + threadIdx.x;   // E * 32 work items
    int e = idx >> 5;
    if (e >= E) return;
    int q = (idx & 31) * 4;
    int s = src[e];
    int d = dst[e];
    float w = dinv[s] * dinv[d];
    const float4 h4 = *(const float4*)&H[s * DIM + q];
    float* a = &agg[d * DIM + q];
    atomicAdd(a + 0, h4.x * w);
    atomicAdd(a + 1, h4.y * w);
    atomicAdd(a + 2, h4.z * w);
    atomicAdd(a + 3, h4.w * w);
}

// out = (agg + H*dinv^2 + b), optional ReLU, optional zeroing of row 0
__global__ void gcn_finalize(const float* __restrict__ agg, const float* __restrict__ H,
                             const float* __restrict__ dinv, const float* __restrict__ bias,
                             float* __restrict__ out, int do_relu, int zero_row0) {
    int idx = blockIdx.x * blockDim.x + threadIdx.x;
    if (idx >= NN * DIM) return;
    int i = idx >> 7;       // node
    int f = idx & 127;      // feature
    float di = dinv[i];
    float v = agg[idx] + H[idx] * di * di + bias[f];
    if (do_relu) v = fmaxf(v, 0.0f);
    if (zero_row0 && i == 0) v = 0.0f;
    out[idx] = v;
}

// ---------------------------------------------------------------------------
// Host side
// ---------------------------------------------------------------------------
static void run_layer(const float* X, const float* W, const float* b,
                      float* H, float* agg,
                      const int* esrc, const int* edst, const float* dinv, int E,
                      float* out, int do_relu, int zero_row0, hipStream_t stream) {
    dim3 blk(256);
    gcn_gemm_wmma<<<dim3((ROW_TILES + 7) / 8), blk, 0, stream>>>(X, W, H);
    gcn_zero_agg<<<dim3((NN * DIM + 255) / 256), blk, 0, stream>>>(agg);
    int work = E * 32;
    gcn_scatter<<<dim3((work + 255) / 256), blk, 0, stream>>>(esrc, edst, dinv, H, agg, E);
    gcn_finalize<<<dim3((NN * DIM + 255) / 256), blk, 0, stream>>>(
        agg, H, dinv, b, out, do_relu, zero_row0);
}

extern "C" void kernel_launch(void* const* d_in, const int* in_sizes, int n_in,
                              void* d_out, int out_size, void* d_ws, size_t ws_size,
                              hipStream_t stream) {
    const float* emb = (const float*)d_in[0];
    const float* W1  = (const float*)d_in[1];
    const float* b1  = (const float*)d_in[2];
    const float* W2  = (const float*)d_in[3];
    const float* b2  = (const float*)d_in[4];
    const float* W3  = (const float*)d_in[5];
    const float* b3  = (const float*)d_in[6];
    const int*   ei  = (const int*)d_in[7];
    const int E = in_sizes[7] / 2;           // edge_index is [2, E] row-major
    const int* esrc = ei;
    const int* edst = ei + E;

    // Workspace layout (floats): deg | dinv | H | agg | X
    float* ws   = (float*)d_ws;
    float* deg  = ws;
    float* dinv = ws + NN;
    float* H    = ws + 2 * NN;
    float* agg  = H + (size_t)NN * DIM;
    float* Xb   = agg + (size_t)NN * DIM;
    float* out  = (float*)d_out;

    dim3 blk(256);
    gcn_init_deg<<<dim3((NN + 255) / 256), blk, 0, stream>>>(deg);
    gcn_accum_deg<<<dim3((E + 255) / 256), blk, 0, stream>>>(edst, deg, E);
    gcn_make_dinv<<<dim3((NN + 255) / 256), blk, 0, stream>>>(deg, dinv);

    // Layer 1: emb -> Xb (ReLU)
    run_layer(emb, W1, b1, H, agg, esrc, edst, dinv, E, Xb, /*relu=*/1, /*zero0=*/0, stream);
    // Layer 2: Xb -> Xb (ReLU); GEMM consumes Xb into H before finalize rewrites Xb
    run_layer(Xb,  W2, b2, H, agg, esrc, edst, dinv, E, Xb, /*relu=*/1, /*zero0=*/0, stream);
    // Layer 3: Xb -> out, no ReLU, zero row 0
    run_layer(Xb,  W3, b3, H, agg, esrc, edst, dinv, E, out, /*relu=*/0, /*zero0=*/1, stream);
}